// MexhatWaveletKanLayer_67499706024691
// MI455X (gfx1250) — compile-verified
//
#include <hip/hip_runtime.h>
#include <hip/hip_bf16.h>

typedef float v2f __attribute__((ext_vector_type(2)));
typedef float v8f __attribute__((ext_vector_type(8)));

#define B_DIM 512
#define I_DIM 512
#define O_DIM 512
#define MEXHAT_NORM 0.8673250705840776f
#define HALF_LOG2E 0.7213475204444817f   /* 0.5 * log2(e) */
#define BN_EPS 1e-5f

#define TB 16   // batch rows per block
#define TO 64   // output cols per block
#define KC 64   // i-chunk staged in LDS

// Async memory->LDS copy, 16B per lane (CDNA5 GLOBAL_LOAD_ASYNC_TO_LDS_B128,
// tracked by ASYNCcnt). LDS address = low 32 bits of the generic pointer
// (ISA 10.2).
__device__ __forceinline__ void async_ld_b128(const float* gptr, const float* lptr) {
  const unsigned int  laddr = (unsigned int)(unsigned long long)(uintptr_t)lptr;
  const unsigned long long gaddr = (unsigned long long)(uintptr_t)gptr;
  asm volatile("global_load_async_to_lds_b128 %0, %1, off"
               :: "v"(laddr), "v"(gaddr)
               : "memory");
}
__device__ __forceinline__ void wait_asynccnt0() {
  asm volatile("s_wait_asynccnt 0x0" ::: "memory");
}

// ---------------------------------------------------------------------------
// Kernel 1: Mexican-hat KAN forward. y[b,o] = sum_i NORM*(t^2-1)*exp(-t^2/2)*w
// with t = (x[b,i]-bias[i,o])*rcp(scale[i,o]).  Transcendental-bound.
// x tiles are double-buffered into LDS with async-to-LDS loads: chunk n+1
// streams in (ASYNCcnt) while v_exp_f32 work runs on chunk n; one barrier
// per chunk.  (s,b,w)[i,o] loads are reused across 4 b-rows per thread and
// stay in L2 (entire problem is ~5 MB, HBM is irrelevant).
// ---------------------------------------------------------------------------
__global__ __launch_bounds__(256) void kan_forward(
    const float* __restrict__ x, const float* __restrict__ scale,
    const float* __restrict__ bias, const float* __restrict__ weight,
    float* __restrict__ Y)
{
  __shared__ float xs[2][TB][KC];

  const int tid = threadIdx.x;
  const int to  = tid & (TO - 1);   // 0..63 (coalesced over o)
  const int tb  = tid >> 6;         // 0..3  (4 b-rows each)
  const int o0  = blockIdx.x * TO;
  const int b0  = blockIdx.y * TB;
  const int o   = o0 + to;

  // staging role: 1 x b128 async load per thread per chunk (16x64 floats)
  const int r_ld = tid >> 4;          // 0..15 row
  const int c_ld = (tid & 15) << 2;   // 0..60 col (float4)

  float acc[4] = {0.f, 0.f, 0.f, 0.f};

  // prime buffer 0
  async_ld_b128(&x[(b0 + r_ld) * I_DIM + c_ld], &xs[0][r_ld][c_ld]);

  int buf = 0;
  for (int i0 = 0; i0 < I_DIM; i0 += KC) {
    wait_asynccnt0();      // this wave's async fill of xs[buf] landed
    __syncthreads();       // everyone's fill landed + prev chunk fully read

    if (i0 + KC < I_DIM)   // prefetch next chunk into the other buffer
      async_ld_b128(&x[(b0 + r_ld) * I_DIM + i0 + KC + c_ld],
                    &xs[buf ^ 1][r_ld][c_ld]);

    #pragma unroll 4
    for (int k = 0; k < KC; ++k) {
      const int  i   = i0 + k;
      const float s  = scale [i * O_DIM + o];
      const float bi = bias  [i * O_DIM + o];
      const float w  = weight[i * O_DIM + o];
      const float inv = __builtin_amdgcn_rcpf(s);  // v_rcp_f32
      const float c0  = -bi * inv;
      const float wn  = w * MEXHAT_NORM;
      #pragma unroll
      for (int r = 0; r < 4; ++r) {
        const float xv = xs[buf][tb * 4 + r][k];    // LDS broadcast per wave
        const float t  = fmaf(xv, inv, c0);
        const float t2 = t * t;
        const float e  = __builtin_amdgcn_exp2f(-HALF_LOG2E * t2); // v_exp_f32
        acc[r] = fmaf(wn, (t2 - 1.0f) * e, acc[r]);
      }
    }
    buf ^= 1;
  }

  #pragma unroll
  for (int r = 0; r < 4; ++r)
    Y[(b0 + tb * 4 + r) * O_DIM + o] = acc[r];
}

// ---------------------------------------------------------------------------
// Kernel 2: BatchNorm statistics via WMMA.  S1[o]=sum_b Y, S2[o]=sum_b Y^2.
// Column sums are a true GEMM: D = A x B with A = Y^T tile (16 o x 4 b) and
// B = ones(4x16), accumulated in the f32 C operand -> v_wmma_f32_16x16x4_f32.
// A-matrix 32-bit layout: lane l<16 holds A[M=l][K=0,1] in v0,v1;
// lanes 16-31 hold K=2,3.  D: lane0 v0..v7 = M 0..7, lane16 v0..v7 = M 8..15.
// ---------------------------------------------------------------------------
__global__ __launch_bounds__(128) void bn_stats_wmma(
    const float* __restrict__ Y, float* __restrict__ S1, float* __restrict__ S2)
{
  const int tid    = threadIdx.x;
  const int lane   = tid & 31;
  const int wave   = tid >> 5;
  const int o_base = (blockIdx.x * 4 + wave) * 16;
  const int l      = lane & 15;
  const int half   = lane >> 4;

  v2f ones; ones.x = 1.0f; ones.y = 1.0f;
  v8f c1 = {}; v8f c2 = {};

  for (int b = 0; b < B_DIM; b += 4) {
    v2f a;
    a.x = Y[(b + 2 * half + 0) * O_DIM + o_base + l];  // A[l][2*half]
    a.y = Y[(b + 2 * half + 1) * O_DIM + o_base + l];  // A[l][2*half+1]
    v2f a2; a2.x = a.x * a.x; a2.y = a.y * a.y;
    c1 = __builtin_amdgcn_wmma_f32_16x16x4_f32(false, a,  false, ones,
                                               (short)0, c1, false, false);
    c2 = __builtin_amdgcn_wmma_f32_16x16x4_f32(false, a2, false, ones,
                                               (short)0, c2, false, false);
  }

  if (l == 0) {  // lanes 0 and 16 each own 8 consecutive o's
    const int ob = o_base + half * 8;
    #pragma unroll
    for (int m = 0; m < 8; ++m) {
      S1[ob + m] = c1[m];
      S2[ob + m] = c2[m];
    }
  }
}

// ---------------------------------------------------------------------------
// Kernel 3: apply BatchNorm (training mode, biased variance).
// ---------------------------------------------------------------------------
__global__ __launch_bounds__(256) void bn_apply(
    const float* __restrict__ Y, const float* __restrict__ S1,
    const float* __restrict__ S2, const float* __restrict__ gamma,
    const float* __restrict__ beta, float* __restrict__ out)
{
  const int idx  = blockIdx.x * 256 + threadIdx.x;
  const int o    = idx & (O_DIM - 1);
  const float mean = S1[o] * (1.0f / (float)B_DIM);
  const float var  = fmaf(-mean, mean, S2[o] * (1.0f / (float)B_DIM));
  const float rstd = __builtin_amdgcn_rsqf(var + BN_EPS);  // v_rsq_f32
  out[idx] = fmaf((Y[idx] - mean) * rstd, gamma[o], beta[o]);
}

// ---------------------------------------------------------------------------
extern "C" void kernel_launch(void* const* d_in, const int* in_sizes, int n_in,
                              void* d_out, int out_size, void* d_ws, size_t ws_size,
                              hipStream_t stream) {
  const float* x      = (const float*)d_in[0];
  const float* scale  = (const float*)d_in[1];
  const float* bias   = (const float*)d_in[2];
  const float* weight = (const float*)d_in[3];
  const float* gamma  = (const float*)d_in[4];
  const float* beta   = (const float*)d_in[5];

  float* Y   = (float*)d_ws;            // B*O floats (1 MB)
  float* S1  = Y + B_DIM * O_DIM;       // O floats
  float* S2  = S1 + O_DIM;              // O floats
  float* out = (float*)d_out;

  dim3 g1(O_DIM / TO, B_DIM / TB);      // 8 x 32 = 256 blocks
  kan_forward<<<g1, 256, 0, stream>>>(x, scale, bias, weight, Y);
  bn_stats_wmma<<<O_DIM / 64, 128, 0, stream>>>(Y, S1, S2);
  bn_apply<<<(B_DIM * O_DIM) / 256, 256, 0, stream>>>(Y, S1, S2, gamma, beta, out);
}